// GinLayer_17583596109847
// MI455X (gfx1250) — compile-verified
//
#include <hip/hip_runtime.h>

#define TP 68    // activation tile pitch in floats (bank-conflict padding)
#define PWP 160  // paired-weight pitch per k2-row (>=128, == 32 mod 64)

typedef __attribute__((ext_vector_type(2))) float v2f;
typedef __attribute__((ext_vector_type(8))) float v8f;

__device__ __forceinline__ v8f wmma4(v2f a, v2f b, v8f c) {
  // D = A(16x4,f32) * B(4x16,f32) + C(16x16,f32)
  return __builtin_amdgcn_wmma_f32_16x16x4_f32(false, a, false, b, (short)0, c,
                                               false, false);
}

// One wave computes a 16x64 tile: tile(16x64 f32, pitch TP) @ W(64x64 f32,
// stored k-pair interleaved: W[(k>>1)*PWP + 2*n + (k&1)]).
// acc[j] holds the 16x16 result tile for cols [16j, 16j+16).
__device__ __forceinline__ void gemm_acc(const float* tile, const float* W,
                                         v8f* acc, int lane) {
  const int m = lane & 15;       // A row / B,D col-in-16
  const int kh = lane >> 4;      // lane-half K sub-offset
#pragma unroll
  for (int s = 0; s < 16; ++s) { // each step covers K += 4
    v2f a = *(const v2f*)&tile[m * TP + 4 * s + 2 * kh];
#pragma unroll
    for (int j = 0; j < 4; ++j) {
      v2f b = *(const v2f*)&W[(2 * s + kh) * PWP + 2 * (16 * j + m)];
      acc[j] = wmma4(a, b, acc[j]);
    }
  }
}

// cooperative load of a 64x64 f32 weight block into LDS, k-pair interleaved
__device__ __forceinline__ void load_w64p(float* dst, const float* src,
                                          int tid) {
#pragma unroll
  for (int i = 0; i < 8; ++i) {
    const int idx = tid + i * 256;  // 0..2047 -> (k2, n)
    const int k2 = idx >> 6, n = idx & 63;
    float2 v;
    v.x = src[(2 * k2) * 64 + n];
    v.y = src[(2 * k2 + 1) * 64 + n];
    *(float2*)&dst[k2 * PWP + 2 * n] = v;
  }
}

// stage 16 contiguous rows [row0, row0+16) of a row-major [*,64] array; OOB -> 0
__device__ __forceinline__ void stage_rows(float* tile, const float* src,
                                           long long row0, long long limit,
                                           int lane) {
  const int r = lane >> 1, p = lane & 1;
  const long long g = row0 + r;
  const float4* s = (const float4*)src + g * 16;
  const float4 z = make_float4(0.f, 0.f, 0.f, 0.f);
#pragma unroll
  for (int i = 0; i < 8; ++i) {
    float4 v = (g < limit) ? s[p * 8 + i] : z;
    *(float4*)&tile[r * TP + (p * 8 + i) * 4] = v;
  }
}

// stage 16 rows of (a + b); OOB -> 0
__device__ __forceinline__ void stage_rows_sum(float* tile, const float* a,
                                               const float* b, long long row0,
                                               long long limit, int lane) {
  const int r = lane >> 1, p = lane & 1;
  const long long g = row0 + r;
  const float4* sa = (const float4*)a + g * 16;
  const float4* sb = (const float4*)b + g * 16;
#pragma unroll
  for (int i = 0; i < 8; ++i) {
    float4 v = make_float4(0.f, 0.f, 0.f, 0.f);
    if (g < limit) {
      float4 x = sa[p * 8 + i];
      float4 y = sb[p * 8 + i];
      v = make_float4(x.x + y.x, x.y + y.y, x.z + y.z, x.w + y.w);
    }
    *(float4*)&tile[r * TP + (p * 8 + i) * 4] = v;
  }
}

// stage 16 gathered rows src[idx[row]]; OOB -> 0
__device__ __forceinline__ void stage_gather(float* tile, const float* src,
                                             const long long* idx,
                                             long long row0, long long limit,
                                             int lane) {
  const int r = lane >> 1, p = lane & 1;
  const long long g = row0 + r;
  const float4 z = make_float4(0.f, 0.f, 0.f, 0.f);
  const long long srow = (g < limit) ? idx[g] : 0;
  const float4* s = (const float4*)src + srow * 16;
#pragma unroll
  for (int i = 0; i < 8; ++i) {
    float4 v = (g < limit) ? s[p * 8 + i] : z;
    *(float4*)&tile[r * TP + (p * 8 + i) * 4] = v;
  }
}

// ---------------- kernel 1: edge message + scatter-add ----------------
// msg = relu(em[src] + ef @ W_le + b_le); aggr[dst] += msg
__global__ __launch_bounds__(256) void k_edge_msg(
    const float* __restrict__ em, const long long* __restrict__ srcI,
    const long long* __restrict__ dstI, const float* __restrict__ ef,
    const float* __restrict__ Wle, const float* __restrict__ ble,
    float* __restrict__ aggr, long long E) {
  __shared__ __align__(16) float sW[32 * PWP];
  __shared__ __align__(16) float sT[8][16 * TP];
  const int tid = threadIdx.x, w = tid >> 5, lane = tid & 31;
  const long long row0 = (long long)blockIdx.x * 128 + (long long)w * 16;

  // prefetch the randomly-gathered em[src] rows so they arrive during the GEMM
  {
    const int r = lane >> 1, p = lane & 1;
    const long long g = row0 + r;
    if (g < E) {
      const long long sr = srcI[g] * 64;
      __builtin_prefetch(&em[sr + p * 32], 0, 3);  // global_prefetch_b8
    }
  }

  load_w64p(sW, Wle, tid);
  stage_rows(sT[w], ef, row0, E, lane);
  __syncthreads();

  v8f acc[4] = {};
  gemm_acc(sT[w], sW, acc, lane);

  const int n = lane & 15, hi = lane >> 4;
  float bias[4];
#pragma unroll
  for (int j = 0; j < 4; ++j) bias[j] = ble[j * 16 + n];
#pragma unroll
  for (int r = 0; r < 8; ++r) {
    const long long g = row0 + hi * 8 + r;
    if (g < E) {
      const long long sr = srcI[g] * 64, dr = dstI[g] * 64;
#pragma unroll
      for (int j = 0; j < 4; ++j) {
        const int col = j * 16 + n;
        float v = acc[j][r] + bias[j] + em[sr + col];
        atomicAdd(&aggr[dr + col], fmaxf(v, 0.f));
      }
    }
  }
}

// ---------------- kernel 2: node MLP (pre-BN) + BN stats ----------------
// h2 = relu((em+aggr)@W1+b1)@W2+b2 -> xout; stats[0:64]+=sum, stats[64:128]+=sumsq
__global__ __launch_bounds__(256) void k_node(
    const float* __restrict__ em, const float* __restrict__ aggr,
    const float* __restrict__ W1, const float* __restrict__ b1,
    const float* __restrict__ W2, const float* __restrict__ b2,
    float* __restrict__ xout, float* __restrict__ stats, long long N) {
  __shared__ __align__(16) float sW[32 * PWP];
  __shared__ __align__(16) float sT[8][16 * TP];
  const int tid = threadIdx.x, w = tid >> 5, lane = tid & 31;
  const long long row0 = (long long)blockIdx.x * 128 + (long long)w * 16;

  load_w64p(sW, W1, tid);
  stage_rows_sum(sT[w], em, aggr, row0, N, lane);
  __syncthreads();

  v8f acc[4] = {};
  gemm_acc(sT[w], sW, acc, lane);

  const int n = lane & 15, hi = lane >> 4;
#pragma unroll
  for (int j = 0; j < 4; ++j) {
    const float bb = b1[j * 16 + n];
#pragma unroll
    for (int r = 0; r < 8; ++r)
      sT[w][(hi * 8 + r) * TP + j * 16 + n] = fmaxf(acc[j][r] + bb, 0.f);
  }
  __syncthreads();
  load_w64p(sW, W2, tid);
  __syncthreads();

  v8f acc2[4] = {};
  gemm_acc(sT[w], sW, acc2, lane);

#pragma unroll
  for (int j = 0; j < 4; ++j) {
    const int col = j * 16 + n;
    const float bb = b2[col];
    float s = 0.f, s2 = 0.f;
#pragma unroll
    for (int r = 0; r < 8; ++r) {
      const long long g = row0 + hi * 8 + r;
      if (g < N) {
        const float v = acc2[j][r] + bb;
        xout[g * 64 + col] = v;
        s += v;
        s2 += v * v;
      }
    }
    atomicAdd(&stats[col], s);
    atomicAdd(&stats[64 + col], s2);
  }
}

// ---------------- kernel 3: edge MLP (pre-BN) + BN stats ----------------
__global__ __launch_bounds__(256) void k_edge_mlp(
    const float* __restrict__ xem, const long long* __restrict__ srcI,
    const long long* __restrict__ dstI, const float* __restrict__ ef,
    const float* __restrict__ We1, const float* __restrict__ be1,
    const float* __restrict__ We2, const float* __restrict__ be2,
    const float* __restrict__ We3, const float* __restrict__ be3,
    float* __restrict__ eout, float* __restrict__ stats, long long E) {
  __shared__ __align__(16) float sW[32 * PWP];
  __shared__ __align__(16) float sT[8][16 * TP];
  const int tid = threadIdx.x, w = tid >> 5, lane = tid & 31;
  const long long row0 = (long long)blockIdx.x * 128 + (long long)w * 16;

  // prefetch the dst-gather rows (used in phase 2) while phase 1 runs
  {
    const int r = lane >> 1, p = lane & 1;
    const long long g = row0 + r;
    if (g < E) {
      const long long dr = dstI[g] * 64;
      __builtin_prefetch(&xem[dr + p * 32], 0, 3);
    }
  }

  v8f acc[4] = {};
  // phase 1: x_em[src] @ We1[0:64]
  load_w64p(sW, We1, tid);
  stage_gather(sT[w], xem, srcI, row0, E, lane);
  __syncthreads();
  gemm_acc(sT[w], sW, acc, lane);
  __syncthreads();
  // phase 2: x_em[dst] @ We1[64:128]
  load_w64p(sW, We1 + 64 * 64, tid);
  stage_gather(sT[w], xem, dstI, row0, E, lane);
  __syncthreads();
  gemm_acc(sT[w], sW, acc, lane);
  __syncthreads();
  // phase 3: ef @ We1[128:192]
  load_w64p(sW, We1 + 128 * 64, tid);
  stage_rows(sT[w], ef, row0, E, lane);
  __syncthreads();
  gemm_acc(sT[w], sW, acc, lane);

  const int n = lane & 15, hi = lane >> 4;
#pragma unroll
  for (int j = 0; j < 4; ++j) {
    const float bb = be1[j * 16 + n];
#pragma unroll
    for (int r = 0; r < 8; ++r)
      sT[w][(hi * 8 + r) * TP + j * 16 + n] = fmaxf(acc[j][r] + bb, 0.f);
  }
  __syncthreads();
  load_w64p(sW, We2, tid);
  __syncthreads();

  v8f acc2[4] = {};
  gemm_acc(sT[w], sW, acc2, lane);
#pragma unroll
  for (int j = 0; j < 4; ++j) {
    const float bb = be2[j * 16 + n];
#pragma unroll
    for (int r = 0; r < 8; ++r)
      sT[w][(hi * 8 + r) * TP + j * 16 + n] = fmaxf(acc2[j][r] + bb, 0.f);
  }
  __syncthreads();
  load_w64p(sW, We3, tid);
  __syncthreads();

  v8f acc3[4] = {};
  gemm_acc(sT[w], sW, acc3, lane);
#pragma unroll
  for (int j = 0; j < 4; ++j) {
    const int col = j * 16 + n;
    const float bb = be3[col];
    float s = 0.f, s2 = 0.f;
#pragma unroll
    for (int r = 0; r < 8; ++r) {
      const long long g = row0 + hi * 8 + r;
      if (g < E) {
        const float v = acc3[j][r] + bb;
        eout[g * 64 + col] = v;
        s += v;
        s2 += v * v;
      }
    }
    atomicAdd(&stats[col], s);
    atomicAdd(&stats[64 + col], s2);
  }
}

// ---------------- small kernels ----------------
__global__ __launch_bounds__(256) void k_zero4(float4* __restrict__ p,
                                               long long n4) {
  long long i = (long long)blockIdx.x * 256 + threadIdx.x;
  if (i < n4) p[i] = make_float4(0.f, 0.f, 0.f, 0.f);
}

// scale/shift from batch stats: scale = g*rsqrt(var+eps), shift = b - mean*scale
__global__ void k_bnparam(const float* __restrict__ stats,
                          const float* __restrict__ gamma,
                          const float* __restrict__ beta,
                          float* __restrict__ ss, float cnt) {
  int c = threadIdx.x;
  if (c < 64) {
    float mean = stats[c] / cnt;
    float var = stats[64 + c] / cnt - mean * mean;
    float sc = gamma[c] * rsqrtf(var + 1e-5f);
    ss[c] = sc;
    ss[64 + c] = beta[c] - mean * sc;
  }
}

// in-place x = relu(x*scale[col] + shift[col]), float4-vectorized
__global__ __launch_bounds__(256) void k_bnapply4(float4* __restrict__ x,
                                                  const float* __restrict__ ss,
                                                  long long total4) {
  long long i = (long long)blockIdx.x * 256 + threadIdx.x;
  if (i < total4) {
    const int c = (int)((i & 15) * 4);  // column of first element
    float4 v = x[i];
    const float4 sc = *(const float4*)&ss[c];
    const float4 sh = *(const float4*)&ss[64 + c];
    v.x = fmaxf(v.x * sc.x + sh.x, 0.f);
    v.y = fmaxf(v.y * sc.y + sh.y, 0.f);
    v.z = fmaxf(v.z * sc.z + sh.z, 0.f);
    v.w = fmaxf(v.w * sc.w + sh.w, 0.f);
    x[i] = v;
  }
}

extern "C" void kernel_launch(void* const* d_in, const int* in_sizes, int n_in,
                              void* d_out, int out_size, void* d_ws,
                              size_t ws_size, hipStream_t stream) {
  const float* em = (const float*)d_in[0];
  const long long* ei = (const long long*)d_in[1];
  const float* ef = (const float*)d_in[2];
  const float* Wle = (const float*)d_in[3];
  const float* ble = (const float*)d_in[4];
  const float* W1 = (const float*)d_in[5];
  const float* b1 = (const float*)d_in[6];
  const float* W2 = (const float*)d_in[7];
  const float* b2 = (const float*)d_in[8];
  const float* gx = (const float*)d_in[9];
  const float* bx = (const float*)d_in[10];
  const float* We1 = (const float*)d_in[11];
  const float* be1 = (const float*)d_in[12];
  const float* We2 = (const float*)d_in[13];
  const float* be2 = (const float*)d_in[14];
  const float* We3 = (const float*)d_in[15];
  const float* be3 = (const float*)d_in[16];
  const float* ge = (const float*)d_in[17];
  const float* beE = (const float*)d_in[18];

  const long long N = (long long)in_sizes[0] / 64;
  const long long E = (long long)in_sizes[2] / 64;
  const long long* srcI = ei;
  const long long* dstI = ei + E;

  float* xout = (float*)d_out;  // [N,64] x_em
  float* eout = xout + N * 64;  // [E,64] edge_out

  float* ws = (float*)d_ws;
  float* aggr = ws;              // [N,64]
  float* nstats = ws + N * 64;   // sum[64], sumsq[64]
  float* estats = nstats + 128;  // sum[64], sumsq[64]
  float* nss = estats + 128;     // scale[64], shift[64]
  float* ess = nss + 128;        // scale[64], shift[64]

  const long long zn4 = (N * 64 + 256) / 4;  // aggr + both stats (contiguous)
  k_zero4<<<(int)((zn4 + 255) / 256), 256, 0, stream>>>((float4*)aggr, zn4);

  const int eblocks = (int)((E + 127) / 128);
  const int nblocks = (int)((N + 127) / 128);

  k_edge_msg<<<eblocks, 256, 0, stream>>>(em, srcI, dstI, ef, Wle, ble, aggr, E);
  k_node<<<nblocks, 256, 0, stream>>>(em, aggr, W1, b1, W2, b2, xout, nstats, N);
  k_bnparam<<<1, 64, 0, stream>>>(nstats, gx, bx, nss, (float)N);
  k_bnapply4<<<(int)((N * 16 + 255) / 256), 256, 0, stream>>>((float4*)xout,
                                                              nss, N * 16);
  k_edge_mlp<<<eblocks, 256, 0, stream>>>(xout, srcI, dstI, ef, We1, be1, We2,
                                          be2, We3, be3, eout, estats, E);
  k_bnparam<<<1, 64, 0, stream>>>(estats, ge, beE, ess, (float)E);
  k_bnapply4<<<(int)((E * 16 + 255) / 256), 256, 0, stream>>>((float4*)eout,
                                                              ess, E * 16);
}